// Yolov1Model_54606214202273
// MI455X (gfx1250) — compile-verified
//
#include <hip/hip_runtime.h>

// ---------------- sizes ----------------
#define BATCH 64
#define KDIM  2304
#define NDIM  1470          // 49 * 30
#define GS    7
#define NCELL 3136          // BATCH * GS * GS
#define NB    2
#define NCLS  20
#define IMGSC 448.0f

typedef __attribute__((ext_vector_type(2))) float v2f;
typedef __attribute__((ext_vector_type(8))) float v8f;

// ---------------------------------------------------------------------------
// Kernel 1: Y = x @ W.T + b  via V_WMMA_F32_16X16X4_F32 (full fp32 fidelity).
// One wave per 16x16 C-tile. grid = (92 N-tiles, 4 M-tiles), block = 32.
// Out-of-range N columns are computed from a clamped (duplicate) W row and
// simply never stored -> branch-free inner loop (no exec-mask churn).
// ---------------------------------------------------------------------------
__global__ __launch_bounds__(32)
void gemm_wmma_f32(const float* __restrict__ X, const float* __restrict__ W,
                   const float* __restrict__ bias, float* __restrict__ Y) {
    const int nt   = blockIdx.x;           // N tile 0..91
    const int mt   = blockIdx.y;           // M tile 0..3
    const int lane = threadIdx.x & 31;
    const int col  = lane & 15;
    const int half = lane >> 4;

    const int m = mt * 16 + col;           // < 64 always
    const int n = nt * 16 + col;
    const bool nok = (n < NDIM);
    const int nc = nok ? n : (NDIM - 1);   // clamp: dead columns, never stored

    // A layout (16x4 f32): lanes 0-15 M=lane {v0:K=0, v1:K=1};
    //                      lanes 16-31 M=lane-16 {v0:K=2, v1:K=3}
    const float* xrow = X + (size_t)m  * KDIM + 2 * half;
    const float* wrow = W + (size_t)nc * KDIM + 2 * half;

    v8f acc = {};
#pragma unroll 8
    for (int k = 0; k < KDIM; k += 4) {
        v2f a, b;
        a.x = xrow[k];
        a.y = xrow[k + 1];
        b.x = wrow[k];
        b.y = wrow[k + 1];
        // 8 args: (neg_a, A, neg_b, B, c_mod, C, reuse_a, reuse_b)
        acc = __builtin_amdgcn_wmma_f32_16x16x4_f32(
            false, a, false, b, (short)0, acc, false, false);
    }

    if (nok) {
        const float bv = bias[n];
#pragma unroll
        for (int r = 0; r < 8; ++r) {
            const int mm = mt * 16 + r + 8 * half;   // D layout: VGPR r -> M=r+8*half
            Y[(size_t)mm * NDIM + n] = acc[r] + bv;
        }
    }
}

// ---------------------------------------------------------------------------
// Kernel 2: per-cell box/score/valid prep (faithful port incl. sc=ch[off+5]).
// ---------------------------------------------------------------------------
__global__ void prep_kernel(const float* __restrict__ Y, float* __restrict__ boxes,
                            float* __restrict__ scores, float* __restrict__ areas,
                            int* __restrict__ valid, int* __restrict__ keep) {
    const int n = blockIdx.x * blockDim.x + threadIdx.x;
    if (n >= NCELL) return;
    const float* ch = Y + (size_t)n * 30;   // (b,gy,gx,30) flattened: n*30+c

    // labels_proba = first-max argmax over 20 class logits, as float
    float best = ch[10];
    int bi = 0;
#pragma unroll
    for (int c = 1; c < NCLS; ++c) {
        const float v = ch[10 + c];
        if (v > best) { best = v; bi = c; }
    }
    const float proba = (float)bi;

    const int gx = n % GS;
    const int gy = (n / GS) % GS;
    const float cell = 1.0f / (float)GS;
    const float nx = (float)gx * cell;
    const float ny = (float)gy * cell;

#pragma unroll
    for (int s = 0; s < NB; ++s) {
        const int off = 5 * s;
        const float b0 = ch[off + 0], b1 = ch[off + 1];
        const float b2 = ch[off + 2], b3 = ch[off + 3];
        const float sc = ch[off + 5];           // faithful: off+5, not off+4
        const int vd = (sc * proba > 0.5f) ? 1 : 0;

        const float cx = b0 * cell + nx;
        const float cy = b1 * cell + ny;
        const float X1 = cx - 0.5f * b2, Y1 = cy - 0.5f * b3;
        const float X2 = cx + 0.5f * b2, Y2 = cy + 0.5f * b3;
        const float bx1 = X1 - 0.5f * X2, by1 = Y1 - 0.5f * Y2;
        const float bx2 = X1 + 0.5f * X2, by2 = Y1 + 0.5f * Y2;

        const int o = s * NCELL + n;
        boxes[(size_t)o * 4 + 0] = bx1;
        boxes[(size_t)o * 4 + 1] = by1;
        boxes[(size_t)o * 4 + 2] = bx2;
        boxes[(size_t)o * 4 + 3] = by2;
        scores[o] = sc;
        areas[o]  = (bx2 - bx1) * (by2 - by1);
        valid[o]  = vd;
        keep[o]   = 0;
    }
}

// ---------------------------------------------------------------------------
// Kernel 3: inverted greedy NMS. One workgroup per slot; sequential select
// loop with LDS argmax reduction; per-thread register-resident box state.
// Early exit when alive set empties (== reference's no-op scan tail).
// ---------------------------------------------------------------------------
#define NMS_T 256
#define MAXE  13            // ceil(3136/256)

__global__ __launch_bounds__(NMS_T)
void nms_kernel(const float* __restrict__ boxes, const float* __restrict__ scores,
                const float* __restrict__ areas, const int* __restrict__ valid,
                int* __restrict__ keep) {
    const int s   = blockIdx.x;
    const int tid = threadIdx.x;
    const float* bx = boxes  + (size_t)s * NCELL * 4;
    const float* sc = scores + (size_t)s * NCELL;
    const float* ar = areas  + (size_t)s * NCELL;
    const int*   vd = valid  + (size_t)s * NCELL;
    int*         kp = keep   + (size_t)s * NCELL;

    float ex1[MAXE], ey1[MAXE], ex2[MAXE], ey2[MAXE], ea[MAXE], es[MAXE];
    bool alive[MAXE], ekeep[MAXE];
#pragma unroll
    for (int j = 0; j < MAXE; ++j) {
        const int n = tid + j * NMS_T;
        ekeep[j] = false;
        if (n < NCELL) {
            ex1[j] = bx[(size_t)n * 4 + 0];
            ey1[j] = bx[(size_t)n * 4 + 1];
            ex2[j] = bx[(size_t)n * 4 + 2];
            ey2[j] = bx[(size_t)n * 4 + 3];
            ea[j]  = ar[n];
            es[j]  = sc[n];
            alive[j] = (vd[n] != 0);
        } else {
            ex1[j] = ey1[j] = ex2[j] = ey2[j] = ea[j] = 0.0f;
            es[j] = -3.0e38f;
            alive[j] = false;
        }
    }

    __shared__ float s_sc[NMS_T];
    __shared__ int   s_ix[NMS_T];
    __shared__ float s_box[5];
    __shared__ int   s_sel;

    for (int iter = 0; iter < NCELL; ++iter) {
        // local argmax over owned alive elements (first-max tie break)
        float lb = -3.0e38f;
        int   li = 0x7fffffff;
#pragma unroll
        for (int j = 0; j < MAXE; ++j) {
            if (alive[j]) {
                const int n = tid + j * NMS_T;
                const float v = es[j];
                if (v > lb || (v == lb && n < li)) { lb = v; li = n; }
            }
        }
        s_sc[tid] = lb;
        s_ix[tid] = li;
        __syncthreads();

        for (int off = NMS_T / 2; off > 0; off >>= 1) {
            if (tid < off) {
                const float v2 = s_sc[tid + off];
                const int   i2 = s_ix[tid + off];
                if (v2 > s_sc[tid] || (v2 == s_sc[tid] && i2 < s_ix[tid])) {
                    s_sc[tid] = v2;
                    s_ix[tid] = i2;
                }
            }
            __syncthreads();
        }

        if (tid == 0) {
            const int sel = s_ix[0];
            if (sel == 0x7fffffff) {
                s_sel = -1;                     // no alive box -> terminate
            } else {
                s_sel = sel;
                s_box[0] = bx[(size_t)sel * 4 + 0];
                s_box[1] = bx[(size_t)sel * 4 + 1];
                s_box[2] = bx[(size_t)sel * 4 + 2];
                s_box[3] = bx[(size_t)sel * 4 + 3];
                s_box[4] = ar[sel];
            }
        }
        __syncthreads();

        const int sel = s_sel;
        if (sel < 0) break;
        const float sx1 = s_box[0], sy1 = s_box[1];
        const float sx2 = s_box[2], sy2 = s_box[3], sa = s_box[4];

#pragma unroll
        for (int j = 0; j < MAXE; ++j) {
            const int n = tid + j * NMS_T;
            if (n >= NCELL) continue;
            if (n == sel) ekeep[j] = true;
            if (alive[j]) {
                const float ix1 = fmaxf(ex1[j], sx1);
                const float iy1 = fmaxf(ey1[j], sy1);
                const float ix2 = fminf(ex2[j], sx2);
                const float iy2 = fminf(ey2[j], sy2);
                float dx = ix2 - ix1; dx = (dx < 0.0f) ? 0.0f : dx;
                float dy = iy2 - iy1; dy = (dy < 0.0f) ? 0.0f : dy;
                const float inter = dx * dy;
                const float iou = inter / (sa + ea[j] - inter);
                alive[j] = (iou >= 0.5f) && (n != sel);
            }
        }
        __syncthreads();   // protect s_sc/s_ix reuse next iteration
    }

#pragma unroll
    for (int j = 0; j < MAXE; ++j) {
        const int n = tid + j * NMS_T;
        if (n < NCELL) kp[n] = ekeep[j] ? 1 : 0;
    }
}

// ---------------------------------------------------------------------------
// Kernel 4: pack output (2, 3136, 25) = [boxes*448, score, 20 labels] * keep
// ---------------------------------------------------------------------------
__global__ void pack_kernel(const float* __restrict__ Y, const float* __restrict__ boxes,
                            const float* __restrict__ scores, const int* __restrict__ keep,
                            float* __restrict__ out) {
    const int t = blockIdx.x * blockDim.x + threadIdx.x;   // over NB*NCELL
    if (t >= NB * NCELL) return;
    const int n = t % NCELL;
    const float k = keep[t] ? 1.0f : 0.0f;
    float* o = out + (size_t)t * 25;
    o[0] = boxes[(size_t)t * 4 + 0] * IMGSC * k;
    o[1] = boxes[(size_t)t * 4 + 1] * IMGSC * k;
    o[2] = boxes[(size_t)t * 4 + 2] * IMGSC * k;
    o[3] = boxes[(size_t)t * 4 + 3] * IMGSC * k;
    o[4] = scores[t] * k;
    const float* lab = Y + (size_t)n * 30 + 10;
#pragma unroll
    for (int c = 0; c < NCLS; ++c) o[5 + c] = lab[c] * k;
}

// ---------------------------------------------------------------------------
extern "C" void kernel_launch(void* const* d_in, const int* in_sizes, int n_in,
                              void* d_out, int out_size, void* d_ws, size_t ws_size,
                              hipStream_t stream) {
    const float* x = (const float*)d_in[0];   // 64*2304
    const float* W = (const float*)d_in[1];   // 1470*2304
    const float* b = (const float*)d_in[2];   // 1470
    float* out = (float*)d_out;               // 2*3136*25

    float* ws     = (float*)d_ws;
    float* Y      = ws;                              // 64*1470 = 94080
    float* boxes  = Y + (size_t)BATCH * NDIM;        // 2*3136*4 = 25088
    float* scores = boxes + (size_t)NB * NCELL * 4;  // 6272
    float* areas  = scores + (size_t)NB * NCELL;     // 6272
    int*   valid  = (int*)(areas + (size_t)NB * NCELL);
    int*   keep   = valid + (size_t)NB * NCELL;

    gemm_wmma_f32<<<dim3(92, 4), 32, 0, stream>>>(x, W, b, Y);
    prep_kernel<<<(NCELL + 255) / 256, 256, 0, stream>>>(Y, boxes, scores, areas, valid, keep);
    nms_kernel<<<NB, NMS_T, 0, stream>>>(boxes, scores, areas, valid, keep);
    pack_kernel<<<(NB * NCELL + 255) / 256, 256, 0, stream>>>(Y, boxes, scores, keep, out);
}